// DAPMMs_46188078301717
// MI455X (gfx1250) — compile-verified
//
#include <hip/hip_runtime.h>

// Problem constants (match reference)
#define Bsz 8
#define Cch 64
#define Nn  9216        // H*W = 96*96
#define Kc  4
#define Dd  129         // 2C+1
#define DP  144         // Dd padded to multiple of 16 (9 tiles)
#define NT  576         // Nn / 16
#define EPS_COV 1e-6f
#define LOG2PI  1.8378770664093453f

typedef __attribute__((ext_vector_type(2))) float v2f;
typedef __attribute__((ext_vector_type(8))) float v8f;

// CDNA5 fp32 WMMA: D(16x16,f32) = A(16x4,f32) x B(4x16,f32) + C
__device__ __forceinline__ v8f wmma4(v2f a, v2f b, v8f c) {
  return __builtin_amdgcn_wmma_f32_16x16x4_f32(false, a, false, b, (short)0, c,
                                               false, false);
}

// ---------------------------------------------------------------------------
// 1. Build z_fg / z_bg  ([B][DP][N], rows >= Dd zeroed)
// ---------------------------------------------------------------------------
__global__ void k_build_z(const float* __restrict__ sup, const float* __restrict__ msk,
                          const float* __restrict__ qry, float* __restrict__ zf,
                          float* __restrict__ zb) {
  int idx = blockIdx.x * blockDim.x + threadIdx.x;
  if (idx >= Bsz * Nn) return;
  int b = idx / Nn, n = idx % Nn;
  float m = msk[(size_t)b * Nn + n];
  const float* sp = sup + (size_t)b * Cch * Nn + n;
  const float* qp = qry + (size_t)b * Cch * Nn + n;
  float ss = 0.f, qq = 0.f, sq = 0.f;
  for (int c = 0; c < Cch; ++c) {
    float s = sp[(size_t)c * Nn], q = qp[(size_t)c * Nn];
    ss += s * s; qq += q * q; sq += s * q;
  }
  float nq  = sqrtf(qq);
  float nfg = m * sqrtf(ss);
  float nbg = (1.f - m) * sqrtf(ss);
  float cf = (m * sq) / (fmaxf(nfg, 1e-8f) * fmaxf(nq, 1e-8f));
  float cb = ((1.f - m) * sq) / (fmaxf(nbg, 1e-8f) * fmaxf(nq, 1e-8f));
  float* zfp = zf + (size_t)b * DP * Nn + n;
  float* zbp = zb + (size_t)b * DP * Nn + n;
  for (int c = 0; c < Cch; ++c) {
    float s = sp[(size_t)c * Nn], q = qp[(size_t)c * Nn];
    zfp[(size_t)c * Nn] = s * m;
    zbp[(size_t)c * Nn] = s * (1.f - m);
    zfp[(size_t)(Cch + c) * Nn] = q;
    zbp[(size_t)(Cch + c) * Nn] = q;
  }
  zfp[(size_t)(2 * Cch) * Nn] = cf;
  zbp[(size_t)(2 * Cch) * Nn] = cb;
  for (int d = Dd; d < DP; ++d) {
    zfp[(size_t)d * Nn] = 0.f;
    zbp[(size_t)d * Nn] = 0.f;
  }
}

// ---------------------------------------------------------------------------
// 2. Pad/transpose W1 (D x 64) -> Wp (64 x DP), zeros for d >= Dd
// ---------------------------------------------------------------------------
__global__ void k_prep_w1(const float* __restrict__ w1, float* __restrict__ wp) {
  int idx = blockIdx.x * blockDim.x + threadIdx.x;
  if (idx >= Cch * DP) return;
  int c = idx / DP, d = idx % DP;
  wp[idx] = (d < Dd) ? w1[d * Cch + c] : 0.f;
}

// ---------------------------------------------------------------------------
// 3. Conv1 via WMMA: h1[b] (64 x N) = Wp (64 x DP) @ z[b] (DP x N) + b1
//    grid (NT, B), block 32 (one wave per 16-column tile; 4 M tiles/wave)
// ---------------------------------------------------------------------------
__global__ void k_conv1_wmma(const float* __restrict__ z, const float* __restrict__ wp,
                             const float* __restrict__ b1, float* __restrict__ h1) {
  int n0 = blockIdx.x * 16;
  int b  = blockIdx.y;
  int lane = threadIdx.x;
  int r = lane & 15, hh = lane >> 4;
  const float* zb = z + (size_t)b * DP * Nn;
  v8f acc[4] = {};
  for (int k0 = 0; k0 < DP; k0 += 4) {
    int k2 = k0 + 2 * hh;
    v2f bf;
    bf[0] = zb[(size_t)k2 * Nn + n0 + r];
    bf[1] = zb[(size_t)(k2 + 1) * Nn + n0 + r];
#pragma unroll
    for (int t = 0; t < 4; ++t) {
      v2f af;
      af[0] = wp[(t * 16 + r) * DP + k2];
      af[1] = wp[(t * 16 + r) * DP + k2 + 1];
      acc[t] = wmma4(af, bf, acc[t]);
    }
  }
#pragma unroll
  for (int t = 0; t < 4; ++t) {
#pragma unroll
    for (int rI = 0; rI < 8; ++rI) {
      int c = t * 16 + rI + 8 * hh;
      h1[((size_t)b * Cch + c) * Nn + n0 + r] = acc[t][rI] + b1[c];
    }
  }
}

// ---------------------------------------------------------------------------
// 4. BatchNorm stats (training mode, biased var) per channel over (B, N)
// ---------------------------------------------------------------------------
__global__ void k_bn_stats(const float* __restrict__ x, int C, float* __restrict__ mean,
                           float* __restrict__ rstd) {
  __shared__ float s1[256], s2[256];
  int c = blockIdx.x, tid = threadIdx.x;
  float a = 0.f, b = 0.f;
  for (int i = tid; i < Bsz * Nn; i += 256) {
    int bb = i / Nn, n = i % Nn;
    float v = x[((size_t)bb * C + c) * Nn + n];
    a += v; b += v * v;
  }
  s1[tid] = a; s2[tid] = b; __syncthreads();
  for (int s = 128; s > 0; s >>= 1) {
    if (tid < s) { s1[tid] += s1[tid + s]; s2[tid] += s2[tid + s]; }
    __syncthreads();
  }
  if (tid == 0) {
    float M = (float)(Bsz * Nn);
    float mu = s1[0] / M;
    float var = s2[0] / M - mu * mu;
    mean[c] = mu;
    rstd[c] = rsqrtf(var + 1e-5f);
  }
}

__global__ void k_bn_relu(float* __restrict__ x, int C, const float* __restrict__ mean,
                          const float* __restrict__ rstd, const float* __restrict__ g,
                          const float* __restrict__ be, int total) {
  int idx = blockIdx.x * blockDim.x + threadIdx.x;
  if (idx >= total) return;
  int c = (idx / Nn) % C;
  float v = (x[idx] - mean[c]) * rstd[c] * g[c] + be[c];
  x[idx] = fmaxf(v, 0.f);
}

// ---------------------------------------------------------------------------
// 5. Conv2 (64 -> 16), VALU (tiny)
// ---------------------------------------------------------------------------
__global__ void k_conv2(const float* __restrict__ h1, const float* __restrict__ w2,
                        const float* __restrict__ b2, float* __restrict__ h2) {
  int idx = blockIdx.x * blockDim.x + threadIdx.x;
  if (idx >= Bsz * 16 * Nn) return;
  int n = idx % Nn, j = (idx / Nn) % 16, b = idx / (Nn * 16);
  const float* xp = h1 + (size_t)b * Cch * Nn + n;
  float s = b2[j];
  for (int c = 0; c < Cch; ++c) s += xp[(size_t)c * Nn] * w2[c * 16 + j];
  h2[idx] = s;
}

// ---------------------------------------------------------------------------
// 6. Conv3 (16 -> K) + softmax over K  -> gamma [B][K][N]
// ---------------------------------------------------------------------------
__global__ void k_conv3_softmax(const float* __restrict__ h2, const float* __restrict__ w3,
                                const float* __restrict__ b3, float* __restrict__ gm) {
  int idx = blockIdx.x * blockDim.x + threadIdx.x;
  if (idx >= Bsz * Nn) return;
  int b = idx / Nn, n = idx % Nn;
  const float* xp = h2 + (size_t)b * 16 * Nn + n;
  float lg[Kc];
#pragma unroll
  for (int k = 0; k < Kc; ++k) lg[k] = b3[k];
  for (int j = 0; j < 16; ++j) {
    float v = xp[(size_t)j * Nn];
#pragma unroll
    for (int k = 0; k < Kc; ++k) lg[k] += v * w3[j * Kc + k];
  }
  float mx = fmaxf(fmaxf(lg[0], lg[1]), fmaxf(lg[2], lg[3]));
  float sum = 0.f;
#pragma unroll
  for (int k = 0; k < Kc; ++k) { lg[k] = expf(lg[k] - mx); sum += lg[k]; }
#pragma unroll
  for (int k = 0; k < Kc; ++k)
    gm[((size_t)b * Kc + k) * Nn + n] = lg[k] / sum;
}

// ---------------------------------------------------------------------------
// 7. sum_g[bk] = sum_n gamma
// ---------------------------------------------------------------------------
__global__ void k_sumg(const float* __restrict__ gm, float* __restrict__ sumg) {
  __shared__ float sm[256];
  int bk = blockIdx.x, tid = threadIdx.x;
  float a = 0.f;
  for (int n = tid; n < Nn; n += 256) a += gm[(size_t)bk * Nn + n];
  sm[tid] = a; __syncthreads();
  for (int s = 128; s > 0; s >>= 1) { if (tid < s) sm[tid] += sm[tid + s]; __syncthreads(); }
  if (tid == 0) sumg[bk] = sm[0];
}

// ---------------------------------------------------------------------------
// 8. mu[bk][d] = sum_n gamma * z / sum_g   (grid: (DP, B*K))
// ---------------------------------------------------------------------------
__global__ void k_mu(const float* __restrict__ z, const float* __restrict__ gm,
                     const float* __restrict__ sumg, float* __restrict__ mu) {
  __shared__ float sm[256];
  int d = blockIdx.x, bk = blockIdx.y, b = bk / Kc, tid = threadIdx.x;
  const float* zp = z + ((size_t)b * DP + d) * Nn;
  const float* gp = gm + (size_t)bk * Nn;
  float a = 0.f;
  for (int n = tid; n < Nn; n += 256) a += gp[n] * zp[n];
  sm[tid] = a; __syncthreads();
  for (int s = 128; s > 0; s >>= 1) { if (tid < s) sm[tid] += sm[tid + s]; __syncthreads(); }
  if (tid == 0) mu[(size_t)bk * DP + d] = sm[0] / sumg[bk];
}

// ---------------------------------------------------------------------------
// 9. Weighted covariance via WMMA:
//    cov[bk] (DP x DP) = (gamma .* diff) (DP x N) @ diff^T (N x DP)
//    grid (81 tiles, B*K), block 32 (one wave per 16x16 output tile)
// ---------------------------------------------------------------------------
__global__ void k_cov_wmma(const float* __restrict__ z, const float* __restrict__ gm,
                           const float* __restrict__ mu, float* __restrict__ cov) {
  int tile = blockIdx.x, bk = blockIdx.y, b = bk / Kc;
  int d0 = (tile / 9) * 16, e0 = (tile % 9) * 16;
  int lane = threadIdx.x, r = lane & 15, hh = lane >> 4;
  const float* za = z + ((size_t)b * DP + d0 + r) * Nn;
  const float* zc = z + ((size_t)b * DP + e0 + r) * Nn;
  const float* gp = gm + (size_t)bk * Nn;
  float muA = mu[(size_t)bk * DP + d0 + r];
  float muB = mu[(size_t)bk * DP + e0 + r];
  v8f acc = {};
  for (int n = 0; n < Nn; n += 4) {
    int n2 = n + 2 * hh;
    if ((n & 127) == 0) {
      __builtin_prefetch(za + n + 512, 0, 1);
      __builtin_prefetch(zc + n + 512, 0, 1);
      __builtin_prefetch(gp + n + 512, 0, 1);
    }
    float g0 = gp[n2], g1 = gp[n2 + 1];
    v2f af, bf;
    af[0] = (za[n2] - muA) * g0;
    af[1] = (za[n2 + 1] - muA) * g1;
    bf[0] = zc[n2] - muB;
    bf[1] = zc[n2 + 1] - muB;
    acc = wmma4(af, bf, acc);
  }
  float* cp = cov + (size_t)bk * DP * DP;
#pragma unroll
  for (int rI = 0; rI < 8; ++rI) {
    int m = d0 + rI + 8 * hh;
    cp[(size_t)m * DP + e0 + r] = acc[rI];
  }
}

// ---------------------------------------------------------------------------
// 10. cov /= sum_g, + eps*I on real diag, identity in padded corner
// ---------------------------------------------------------------------------
__global__ void k_cov_fixup(float* __restrict__ cov, const float* __restrict__ sumg) {
  int idx = blockIdx.x * blockDim.x + threadIdx.x;
  if (idx >= Bsz * Kc * DP * DP) return;
  int bk = idx / (DP * DP), rem = idx % (DP * DP);
  int d = rem / DP, e = rem % DP;
  float v = cov[idx] / sumg[bk];
  if (d == e) v = (d < Dd) ? v + EPS_COV : 1.f;
  cov[idx] = v;
}

// ---------------------------------------------------------------------------
// 11. In-place Cholesky (lower), zero upper, logdet. One block per (b,k).
// ---------------------------------------------------------------------------
__global__ void k_cholesky(float* __restrict__ cov, float* __restrict__ logdet) {
  int bk = blockIdx.x, i = threadIdx.x;
  float* M = cov + (size_t)bk * DP * DP;
  for (int j = 0; j < DP; ++j) {
    if (i == j) {
      float s = M[(size_t)j * DP + j];
      for (int p = 0; p < j; ++p) { float l = M[(size_t)j * DP + p]; s -= l * l; }
      M[(size_t)j * DP + j] = sqrtf(fmaxf(s, 1e-20f));
    }
    __syncthreads();
    if (i > j && i < DP) {
      float s = M[(size_t)i * DP + j];
      for (int p = 0; p < j; ++p) s -= M[(size_t)i * DP + p] * M[(size_t)j * DP + p];
      M[(size_t)i * DP + j] = s / M[(size_t)j * DP + j];
    } else if (i < j) {
      M[(size_t)i * DP + j] = 0.f;
    }
    __syncthreads();
  }
  if (i == 0) {
    float s = 0.f;
    for (int d = 0; d < Dd; ++d) s += logf(M[(size_t)d * DP + d]);
    logdet[bk] = 2.f * s;
  }
}

// ---------------------------------------------------------------------------
// 12. Lower-triangular inverse: linv = L^-1. Thread c solves column c.
// ---------------------------------------------------------------------------
__global__ void k_trinv(const float* __restrict__ cov, float* __restrict__ linv) {
  int bk = blockIdx.x, c = threadIdx.x;
  if (c >= DP) return;
  const float* L = cov + (size_t)bk * DP * DP;
  float* X = linv + (size_t)bk * DP * DP;
  for (int i = 0; i < c; ++i) X[(size_t)i * DP + c] = 0.f;
  for (int i = c; i < DP; ++i) {
    float s = (i == c) ? 1.f : 0.f;
    for (int p = c; p < i; ++p) s -= L[(size_t)i * DP + p] * X[(size_t)p * DP + c];
    X[(size_t)i * DP + c] = s / L[(size_t)i * DP + i];
  }
}

// ---------------------------------------------------------------------------
// 13. quad[bk][n] = || L^-1 (z - mu) ||^2 via WMMA:
//     Y (DP x 16) = linv (DP x DP) @ diff (DP x 16); column sum of squares.
//     grid (NT, B*K), block 32. 9 accumulators per wave.
// ---------------------------------------------------------------------------
__global__ void k_quad_wmma(const float* __restrict__ z, const float* __restrict__ mu,
                            const float* __restrict__ linv, float* __restrict__ quad) {
  int n0 = blockIdx.x * 16;
  int bk = blockIdx.y, b = bk / Kc;
  int lane = threadIdx.x, r = lane & 15, hh = lane >> 4;
  const float* zb = z + (size_t)b * DP * Nn;
  const float* mp = mu + (size_t)bk * DP;
  const float* Lv = linv + (size_t)bk * DP * DP;
  v8f acc[9] = {};
  for (int k0 = 0; k0 < DP; k0 += 4) {
    int k2 = k0 + 2 * hh;
    v2f bf;
    bf[0] = zb[(size_t)k2 * Nn + n0 + r] - mp[k2];
    bf[1] = zb[(size_t)(k2 + 1) * Nn + n0 + r] - mp[k2 + 1];
#pragma unroll
    for (int t = 0; t < 9; ++t) {
      v2f af;
      af[0] = Lv[(size_t)(t * 16 + r) * DP + k2];
      af[1] = Lv[(size_t)(t * 16 + r) * DP + k2 + 1];
      acc[t] = wmma4(af, bf, acc[t]);
    }
  }
  float s = 0.f;
#pragma unroll
  for (int t = 0; t < 9; ++t)
#pragma unroll
    for (int rI = 0; rI < 8; ++rI) s += acc[t][rI] * acc[t][rI];
  s += __shfl_xor(s, 16);
  if (lane < 16) quad[(size_t)bk * Nn + n0 + lane] = s;
}

// ---------------------------------------------------------------------------
// 14. Sample energy: -mean(logsumexp_k logp). Two-stage deterministic reduce.
// ---------------------------------------------------------------------------
__global__ void k_energy1(const float* __restrict__ quad, const float* __restrict__ logdet,
                          const float* __restrict__ sumg, float* __restrict__ partial) {
  __shared__ float sm[256];
  int tid = threadIdx.x;
  int idx = blockIdx.x * 256 + tid;  // exactly B*N work items
  int b = idx / Nn, n = idx % Nn;
  float lp[Kc];
#pragma unroll
  for (int k = 0; k < Kc; ++k) {
    int bk = b * Kc + k;
    float phi = sumg[bk] / (float)Nn;
    lp[k] = -0.5f * quad[(size_t)bk * Nn + n]
            - 0.5f * (logdet[bk] + (float)Dd * LOG2PI) + logf(phi + 1e-12f);
  }
  float mx = fmaxf(fmaxf(lp[0], lp[1]), fmaxf(lp[2], lp[3]));
  float e = expf(lp[0] - mx) + expf(lp[1] - mx) + expf(lp[2] - mx) + expf(lp[3] - mx);
  sm[tid] = mx + logf(e);
  __syncthreads();
  for (int s = 128; s > 0; s >>= 1) { if (tid < s) sm[tid] += sm[tid + s]; __syncthreads(); }
  if (tid == 0) partial[blockIdx.x] = sm[0];
}

__global__ void k_energy2(const float* __restrict__ partial, int nblk, float* __restrict__ outp) {
  __shared__ float sm[256];
  int tid = threadIdx.x;
  float a = 0.f;
  for (int i = tid; i < nblk; i += 256) a += partial[i];
  sm[tid] = a; __syncthreads();
  for (int s = 128; s > 0; s >>= 1) { if (tid < s) sm[tid] += sm[tid + s]; __syncthreads(); }
  if (tid == 0) outp[0] = -(sm[0] / (float)(Bsz * Nn));
}

// ---------------------------------------------------------------------------
// 15. fs[bk][d] = (mu + L @ noise) * phi
// ---------------------------------------------------------------------------
__global__ void k_sample(const float* __restrict__ mu, const float* __restrict__ L,
                         const float* __restrict__ sumg, const float* __restrict__ noise,
                         float* __restrict__ fs) {
  int idx = blockIdx.x * blockDim.x + threadIdx.x;
  if (idx >= Bsz * Kc * DP) return;
  int bk = idx / DP, d = idx % DP;
  if (d >= Dd) { fs[idx] = 0.f; return; }
  const float* Lr = L + ((size_t)bk * DP + d) * DP;
  const float* np = noise + (size_t)bk * Dd;
  float s = mu[(size_t)bk * DP + d];
  for (int e = 0; e <= d; ++e) s += Lr[e] * np[e];
  fs[idx] = s * (sumg[bk] / (float)Nn);
}

// ---------------------------------------------------------------------------
// 16. mu_stack output: [K][B][D]
// ---------------------------------------------------------------------------
__global__ void k_mustack(const float* __restrict__ fsf, float* __restrict__ out) {
  int idx = blockIdx.x * blockDim.x + threadIdx.x;
  if (idx >= Kc * Bsz * Dd) return;
  int k = idx / (Bsz * Dd), b = (idx / Dd) % Bsz, d = idx % Dd;
  out[idx] = fsf[((size_t)b * Kc + k) * DP + d];
}

// ---------------------------------------------------------------------------
// 17. Discriminative softmax over 2K prototypes -> Prob_map [B][2][H][W]
// ---------------------------------------------------------------------------
__global__ void k_probmap(const float* __restrict__ zf, const float* __restrict__ fsf,
                          const float* __restrict__ fsb, float* __restrict__ out) {
  int idx = blockIdx.x * blockDim.x + threadIdx.x;
  if (idx >= Bsz * Nn) return;
  int b = idx / Nn, n = idx % Nn;
  const float* zp = zf + (size_t)b * DP * Nn + n;
  float lg[2 * Kc] = {};
  for (int d = 0; d < Dd; ++d) {
    float zv = zp[(size_t)d * Nn];
#pragma unroll
    for (int m = 0; m < Kc; ++m) {
      lg[m]      += zv * fsf[((size_t)b * Kc + m) * DP + d];
      lg[Kc + m] += zv * fsb[((size_t)b * Kc + m) * DP + d];
    }
  }
  float mx = lg[0];
#pragma unroll
  for (int m = 1; m < 2 * Kc; ++m) mx = fmaxf(mx, lg[m]);
  float sum = 0.f;
#pragma unroll
  for (int m = 0; m < 2 * Kc; ++m) { lg[m] = expf(lg[m] - mx); sum += lg[m]; }
  float pf = (lg[0] + lg[1] + lg[2] + lg[3]) / sum;
  float pb = (lg[4] + lg[5] + lg[6] + lg[7]) / sum;
  out[((size_t)b * 2 + 0) * Nn + n] = pb;
  out[((size_t)b * 2 + 1) * Nn + n] = pf;
}

// ---------------------------------------------------------------------------
extern "C" void kernel_launch(void* const* d_in, const int* in_sizes, int n_in,
                              void* d_out, int out_size, void* d_ws, size_t ws_size,
                              hipStream_t stream) {
  const float* sup = (const float*)d_in[0];
  const float* msk = (const float*)d_in[1];
  const float* qry = (const float*)d_in[2];
  const float* w1  = (const float*)d_in[3];
  const float* b1  = (const float*)d_in[4];
  const float* g1  = (const float*)d_in[5];
  const float* be1 = (const float*)d_in[6];
  const float* w2  = (const float*)d_in[7];
  const float* b2  = (const float*)d_in[8];
  const float* g2  = (const float*)d_in[9];
  const float* be2 = (const float*)d_in[10];
  const float* w3  = (const float*)d_in[11];
  const float* b3  = (const float*)d_in[12];
  const float* noise[2] = {(const float*)d_in[13], (const float*)d_in[14]};
  float* out = (float*)d_out;

  float* w = (float*)d_ws;
  size_t off = 0;
  auto alloc = [&](size_t nel) { float* p = w + off; off += nel; return p; };
  float* z[2]; z[0] = alloc((size_t)Bsz * DP * Nn); z[1] = alloc((size_t)Bsz * DP * Nn);
  float* wp1  = alloc((size_t)Cch * DP);
  float* h1   = alloc((size_t)Bsz * Cch * Nn);
  float* h2   = alloc((size_t)Bsz * 16 * Nn);
  float* gm   = alloc((size_t)Bsz * Kc * Nn);
  float* sumg = alloc(Bsz * Kc);
  float* muv  = alloc(Bsz * Kc * DP);
  float* cov  = alloc((size_t)Bsz * Kc * DP * DP);
  float* lin  = alloc((size_t)Bsz * Kc * DP * DP);
  float* qd   = alloc((size_t)Bsz * Kc * Nn);
  float* ld   = alloc(Bsz * Kc);
  float* fs[2]; fs[0] = alloc(Bsz * Kc * DP); fs[1] = alloc(Bsz * Kc * DP);
  float* bn1m = alloc(Cch); float* bn1r = alloc(Cch);
  float* bn2m = alloc(16);  float* bn2r = alloc(16);
  float* part = alloc(512);

  const int eoff = Kc * Bsz * Dd + Bsz * 2 * Nn;  // 4128 + 147456

  k_build_z<<<(Bsz * Nn + 255) / 256, 256, 0, stream>>>(sup, msk, qry, z[0], z[1]);
  k_prep_w1<<<(Cch * DP + 255) / 256, 256, 0, stream>>>(w1, wp1);

  for (int br = 0; br < 2; ++br) {
    k_conv1_wmma<<<dim3(NT, Bsz), 32, 0, stream>>>(z[br], wp1, b1, h1);
    k_bn_stats<<<Cch, 256, 0, stream>>>(h1, Cch, bn1m, bn1r);
    k_bn_relu<<<(Bsz * Cch * Nn + 255) / 256, 256, 0, stream>>>(h1, Cch, bn1m, bn1r, g1, be1,
                                                               Bsz * Cch * Nn);
    k_conv2<<<(Bsz * 16 * Nn + 255) / 256, 256, 0, stream>>>(h1, w2, b2, h2);
    k_bn_stats<<<16, 256, 0, stream>>>(h2, 16, bn2m, bn2r);
    k_bn_relu<<<(Bsz * 16 * Nn + 255) / 256, 256, 0, stream>>>(h2, 16, bn2m, bn2r, g2, be2,
                                                              Bsz * 16 * Nn);
    k_conv3_softmax<<<(Bsz * Nn + 255) / 256, 256, 0, stream>>>(h2, w3, b3, gm);
    k_sumg<<<Bsz * Kc, 256, 0, stream>>>(gm, sumg);
    k_mu<<<dim3(DP, Bsz * Kc), 256, 0, stream>>>(z[br], gm, sumg, muv);
    k_cov_wmma<<<dim3(81, Bsz * Kc), 32, 0, stream>>>(z[br], gm, muv, cov);
    k_cov_fixup<<<(Bsz * Kc * DP * DP + 255) / 256, 256, 0, stream>>>(cov, sumg);
    k_cholesky<<<Bsz * Kc, 160, 0, stream>>>(cov, ld);
    k_trinv<<<Bsz * Kc, 160, 0, stream>>>(cov, lin);
    k_quad_wmma<<<dim3(NT, Bsz * Kc), 32, 0, stream>>>(z[br], muv, lin, qd);
    k_energy1<<<(Bsz * Nn) / 256, 256, 0, stream>>>(qd, ld, sumg, part);
    k_energy2<<<1, 256, 0, stream>>>(part, (Bsz * Nn) / 256, out + eoff + br);
    k_sample<<<(Bsz * Kc * DP + 255) / 256, 256, 0, stream>>>(muv, cov, sumg, noise[br], fs[br]);
  }

  k_mustack<<<(Kc * Bsz * Dd + 255) / 256, 256, 0, stream>>>(fs[0], out);
  k_probmap<<<(Bsz * Nn + 255) / 256, 256, 0, stream>>>(z[0], fs[0], fs[1], out + Kc * Bsz * Dd);
}